// GNNModule_13786845020235
// MI455X (gfx1250) — compile-verified
//
#include <hip/hip_runtime.h>
#include <hip/hip_bf16.h>

#define EPSV 1e-5f

typedef __attribute__((ext_vector_type(2))) float v2f;
typedef __attribute__((ext_vector_type(8))) float v8f;

// ---------------------------------------------------------------------------
// zero (vectorized): element count must be a multiple of 4 (all buffers are)
// ---------------------------------------------------------------------------
__global__ __launch_bounds__(256) void zero4_kernel(float4* __restrict__ p, unsigned n4) {
    unsigned i = blockIdx.x * blockDim.x + threadIdx.x;
    if (i < n4) p[i] = make_float4(0.f, 0.f, 0.f, 0.f);
}

// ---------------------------------------------------------------------------
// spmm: out[dst[e]] += zin[sidx]  (16 floats per row, float4 per thread)
//   sidx = src ? src[e] : e;  if (gidx) sidx = gidx[sidx]   (fused gather)
// ---------------------------------------------------------------------------
__global__ __launch_bounds__(256)
void spmm_kernel(const float* __restrict__ zin, const int* __restrict__ src,
                 const int* __restrict__ gidx, const int* __restrict__ dst,
                 float* __restrict__ zout, unsigned nE4) {
    unsigned t = blockIdx.x * blockDim.x + threadIdx.x;
    if (t >= nE4) return;
    unsigned e = t >> 2;
    unsigned c = (t & 3u) * 4u;
    unsigned s = src ? (unsigned)src[e] : e;
    if (gidx) s = (unsigned)gidx[s];
    unsigned d = (unsigned)dst[e];
    float4 v = *(const float4*)(zin + s * 16u + c);
    float* o = zout + d * 16u + c;
    atomicAdd(o + 0, v.x);
    atomicAdd(o + 1, v.y);
    atomicAdd(o + 2, v.z);
    atomicAdd(o + 3, v.w);
}

__device__ __forceinline__ v8f wmma4(v2f a, v2f b, v8f c) {
    return __builtin_amdgcn_wmma_f32_16x16x4_f32(false, a, false, b, (short)0, c,
                                                 false, false);
}

// ---------------------------------------------------------------------------
// Fused K=96 GEMM + bias + ReLU-combine + BN statistics.
//   h(r, 0:32) = sum_{s=0..5} (scale_s(r) * z_s[r, :]) @ W_s^T + bias_s
//   t(r, c)    = h(r, c) + relu(h(r, c+16)),  c in 0..15
// One wave32 per 16-row tile; 24 K-steps x 2 column tiles = 48 WMMAs (two
// interleaved accumulation chains). The lane that owns column c of tile nt=0
// owns column c+16 of tile nt=1 for the SAME rows, so t is formed in-register.
// t is written column-major [16][stride] (stride = rows padded to 16) as two
// float4 stores per lane. Column sum/sumsq: in-lane 8-row partial, shfl_xor
// pair-reduce, LDS ds_add, then 32 global atomics per block.
// ---------------------------------------------------------------------------
__global__ __launch_bounds__(256)
void fused6_gemm_bn_kernel(const float* __restrict__ z0, const float* __restrict__ z1,
                           const float* __restrict__ z2, const float* __restrict__ z3,
                           const float* __restrict__ z4, const float* __restrict__ z5,
                           const float* __restrict__ deg,    // scales segment 1 rows
                           const float* __restrict__ W6,     // [6][32][16]
                           const float* __restrict__ bias6,  // [6][32]
                           float* __restrict__ t,            // col-major [16][stride]
                           float* __restrict__ stats,        // sum[16] | sumsq[16]
                           unsigned nrows, unsigned stride) {
    __shared__ float ssum[16];
    __shared__ float ssq[16];
    const unsigned tid = threadIdx.x;
    if (tid < 16u) { ssum[tid] = 0.0f; ssq[tid] = 0.0f; }
    __syncthreads();

    const unsigned lane = tid & 31u;
    const unsigned waveInBlk = tid >> 5;
    const unsigned r0 = (blockIdx.x * 8u + waveInBlk) * 16u;
    const unsigned hi = lane >> 4;    // 0: lanes 0-15, 1: lanes 16-31
    const unsigned l15 = lane & 15u;

    if (r0 < nrows) {                 // wave-uniform; EXEC stays full inside
        unsigned row = r0 + l15;
        if (row >= nrows) row = nrows - 1u;   // clamp loads, keep EXEC full
        const unsigned rbase = r0 + hi * 8u;  // this lane's first of 8 rows
        const float ds = deg[row];

        __builtin_prefetch(z0 + (row + 2048u) * 16u, 0, 0);

        // A tiles: 6 segments x 4 K-chunks. 16x4 f32 layout:
        // lanes 0-15 hold K=kb,kb+1 ; lanes 16-31 hold K=kb+2,kb+3.
        const float* zp[6] = {z0, z1, z2, z3, z4, z5};
        v2f a[6][4];
#pragma unroll
        for (int s = 0; s < 6; ++s) {
            const float* zr = zp[s] + row * 16u;
            const float sc = (s == 1) ? ds : 1.0f;
#pragma unroll
            for (int kk = 0; kk < 4; ++kk) {
                unsigned kb = (unsigned)kk * 4u + hi * 2u;
                a[s][kk][0] = zr[kb + 0u] * sc;
                a[s][kk][1] = zr[kb + 1u] * sc;
            }
        }

        v8f c0, c1;
#pragma unroll
        for (int v = 0; v < 8; ++v) { c0[v] = 0.0f; c1[v] = 0.0f; }

#pragma unroll
        for (int s = 0; s < 6; ++s) {
            const float* w0 = W6 + (unsigned)s * 512u + l15 * 16u;          // cols 0..15
            const float* w1 = W6 + (unsigned)s * 512u + (16u + l15) * 16u;  // cols 16..31
#pragma unroll
            for (int kk = 0; kk < 4; ++kk) {
                unsigned kb = (unsigned)kk * 4u + hi * 2u;
                v2f b0, b1;
                b0[0] = w0[kb + 0u]; b0[1] = w0[kb + 1u];
                b1[0] = w1[kb + 0u]; b1[1] = w1[kb + 1u];
                c0 = wmma4(a[s][kk], b0, c0);   // two independent chains -> ILP
                c1 = wmma4(a[s][kk], b1, c1);
            }
        }

        // epilogue: combined bias, relu-combine, t store, stat partials
        float bl = 0.0f, br = 0.0f;
#pragma unroll
        for (int j = 0; j < 6; ++j) {
            bl += bias6[j * 32 + (int)l15];
            br += bias6[j * 32 + 16 + (int)l15];
        }
        float tv[8];
        float s8 = 0.0f, q8 = 0.0f;
#pragma unroll
        for (int v = 0; v < 8; ++v) {
            float right = c1[v] + br;
            float val = (c0[v] + bl) + (right > 0.0f ? right : 0.0f);
            tv[v] = val;
            if (rbase + (unsigned)v < nrows) { s8 += val; q8 += val * val; }
        }
        float* tc = t + l15 * stride + rbase;   // stride % 16 == 0 -> aligned
        *(float4*)(tc)     = make_float4(tv[0], tv[1], tv[2], tv[3]);
        *(float4*)(tc + 4) = make_float4(tv[4], tv[5], tv[6], tv[7]);

        s8 += __shfl_xor(s8, 16, 32);   // combine the two half-wave row groups
        q8 += __shfl_xor(q8, 16, 32);
        if (hi == 0u) {
            atomicAdd(&ssum[l15], s8);
            atomicAdd(&ssq[l15], q8);
        }
    }
    __syncthreads();
    if (tid < 16u) {
        atomicAdd(&stats[tid], ssum[tid]);
        atomicAdd(&stats[16u + tid], ssq[tid]);
    }
}

// ---------------------------------------------------------------------------
// BN finalize: out(r, c) = (t(c, r) - m_c) * rsqrt(v_c + eps) * w_c + b_c
// One thread per row: 16 coalesced column loads, 4 float4 row stores.
// ---------------------------------------------------------------------------
__global__ __launch_bounds__(256)
void bnfin_kernel(const float* __restrict__ t, const float* __restrict__ stats,
                  const float* __restrict__ w, const float* __restrict__ b,
                  float* __restrict__ out, unsigned n, unsigned stride) {
    __shared__ float sscale[16];
    __shared__ float sshift[16];
    const unsigned tid = threadIdx.x;
    if (tid < 16u) {
        float invn = 1.0f / (float)n;
        float m = stats[tid] * invn;
        float var = stats[16u + tid] * invn - m * m;
        float sc = rsqrtf(var + EPSV) * w[tid];
        sscale[tid] = sc;
        sshift[tid] = b[tid] - m * sc;
    }
    __syncthreads();

    unsigned r = blockIdx.x * blockDim.x + tid;
    if (r >= n) return;
    const float* tr = t + r;
    float o[16];
#pragma unroll
    for (int c = 0; c < 16; ++c)
        o[c] = tr[(size_t)c * stride] * sscale[c] + sshift[c];
    float4* orow = (float4*)(out + (size_t)r * 16u);
#pragma unroll
    for (int q = 0; q < 4; ++q)
        orow[q] = make_float4(o[4 * q], o[4 * q + 1], o[4 * q + 2], o[4 * q + 3]);
}

// ---------------------------------------------------------------------------
static inline unsigned nblk(size_t n, unsigned b) { return (unsigned)((n + b - 1) / b); }

extern "C" void kernel_launch(void* const* d_in, const int* in_sizes, int n_in,
                              void* d_out, int out_size, void* d_ws, size_t ws_size,
                              hipStream_t stream) {
    (void)n_in; (void)out_size; (void)ws_size;
    const float* x      = (const float*)d_in[0];
    const float* y      = (const float*)d_in[1];
    const float* deg_g  = (const float*)d_in[2];
    const float* deg_lg = (const float*)d_in[3];
    const float* thW    = (const float*)d_in[4];   // [6][32][16]
    const float* thB    = (const float*)d_in[5];   // [6][32]
    const float* gmW    = (const float*)d_in[6];
    const float* gmB    = (const float*)d_in[7];
    const float* bnxw   = (const float*)d_in[8];
    const float* bnxb   = (const float*)d_in[9];
    const float* bnyw   = (const float*)d_in[10];
    const float* bnyb   = (const float*)d_in[11];
    const int* src_g    = (const int*)d_in[12];
    const int* dst_g    = (const int*)d_in[13];
    const int* src_lg   = (const int*)d_in[14];
    const int* dst_lg   = (const int*)d_in[15];
    const int* eid2nid  = (const int*)d_in[16];

    const size_t N  = (size_t)in_sizes[2];    // deg_g has N elements
    const size_t E  = (size_t)in_sizes[3];    // deg_lg has E elements
    const size_t EL = (size_t)in_sizes[14];   // src_lg has EL elements
    const size_t Np = (N + 15) & ~(size_t)15; // padded row counts for t buffers
    const size_t Ep = (E + 15) & ~(size_t)15;

    // workspace layout
    float* ws = (float*)d_ws;
    size_t o = 0;
    float* n1   = ws + o; o += N * 16;   // A x
    float* n2   = ws + o; o += N * 16;   // A^2 x
    float* n3   = ws + o; o += N * 16;   // A^4 x
    float* ntmp = ws + o; o += N * 16;   // A^3 x, then yx
    float* tx   = ws + o; o += Np * 16;  // node t (col-major, padded)
    float* e1   = ws + o; o += E * 16;   // L y
    float* e2   = ws + o; o += E * 16;   // L^2 y
    float* e3   = ws + o; o += E * 16;   // L^4 y
    float* etmp = ws + o; o += E * 16;   // L^3 y, then xy_agg
    float* ty   = ws + o; o += Ep * 16;  // edge t (col-major, padded)
    float* stats = ws + o; o += 64;      // [sum_x, sq_x | sum_y, sq_y] x16

    const int* NOIDX = nullptr;

#define ZERO(p, n)  zero4_kernel<<<nblk((n) / 4, 256), 256, 0, stream>>>((float4*)(p), (unsigned)((n) / 4))
#define SPMM(zi, s, g, d, zo, ne) \
    spmm_kernel<<<nblk((ne) * 4, 256), 256, 0, stream>>>((zi), (s), (g), (d), (zo), (unsigned)((ne) * 4))

    ZERO(stats, (size_t)64);

    // ---------------- node branch ----------------
    ZERO(n1,   N * 16); SPMM(x,    src_g, NOIDX, dst_g, n1,   E);  // A x
    ZERO(n2,   N * 16); SPMM(n1,   src_g, NOIDX, dst_g, n2,   E);  // A^2 x
    ZERO(ntmp, N * 16); SPMM(n2,   src_g, NOIDX, dst_g, ntmp, E);  // A^3 x
    ZERO(n3,   N * 16); SPMM(ntmp, src_g, NOIDX, dst_g, n3,   E);  // A^4 x
    ZERO(ntmp, N * 16); SPMM(y,    NOIDX, NOIDX, dst_g, ntmp, E);  // yx (copy_edge+sum)
    fused6_gemm_bn_kernel<<<nblk(N, 128), 256, 0, stream>>>(
        x, x, ntmp, n1, n2, n3, deg_g, thW, thB, tx, stats,
        (unsigned)N, (unsigned)Np);
    bnfin_kernel<<<nblk(N, 256), 256, 0, stream>>>(
        tx, stats, bnxw, bnxb, (float*)d_out, (unsigned)N, (unsigned)Np);

    // ---------------- line-graph (edge) branch ----------------
    ZERO(e1,   E * 16); SPMM(y,    src_lg, NOIDX,   dst_lg, e1,   EL);  // L y
    ZERO(e2,   E * 16); SPMM(e1,   src_lg, NOIDX,   dst_lg, e2,   EL);  // L^2 y
    ZERO(etmp, E * 16); SPMM(e2,   src_lg, NOIDX,   dst_lg, etmp, EL);  // L^3 y
    ZERO(e3,   E * 16); SPMM(etmp, src_lg, NOIDX,   dst_lg, e3,   EL);  // L^4 y
    ZERO(etmp, E * 16); SPMM(x,    src_lg, eid2nid, dst_lg, etmp, EL);  // xy_agg (fused gather)
    fused6_gemm_bn_kernel<<<nblk(E, 128), 256, 0, stream>>>(
        y, y, etmp, e1, e2, e3, deg_lg, gmW, gmB, ty, stats + 32,
        (unsigned)E, (unsigned)Ep);
    bnfin_kernel<<<nblk(E, 256), 256, 0, stream>>>(
        ty, stats + 32, bnyw, bnyb, (float*)d_out + N * 16, (unsigned)E, (unsigned)Ep);

#undef ZERO
#undef SPMM
}